// Upsampling_29171417874896
// MI455X (gfx1250) — compile-verified
//
#include <hip/hip_runtime.h>

// Nearest-neighbor 2x upsample, fp32.
//   in : (8, 64, 256, 256)  = 512 planes of 256x256
//   out: (8, 64, 512, 512)  = 512 planes of 512x512
//
// Output-centric: thread t owns 8 consecutive output floats (two float4s in
// one output row) = 4 input floats duplicated. One global_load_b128, two
// non-temporal global_store_b128.
//
// All addressing is done as unsigned 32-bit *float4-element* indices through
// v4f pointers so the backend can use the gfx1250 GVS saddr + voffset
// scale_offset form (base SGPR pair + vgpr index scaled by 16 in hardware)
// instead of materializing 64-bit addresses in VALU.
//
// Roofline: 128 MiB read + 512 MiB written -> ~27 us at 23.3 TB/s. Each input
// row feeds two output rows; the whole input fits in the 192 MB L2, so the
// second pass is an L2 hit as long as the output stream doesn't evict it —
// hence TH_STORE_NT on the write-once output.

typedef float v4f __attribute__((ext_vector_type(4)));

__global__ __launch_bounds__(256) void
upsample2x_nn_f32_kernel(const v4f* __restrict__ in4,
                         v4f* __restrict__ out4,
                         unsigned n8) {
    unsigned tid = blockIdx.x * blockDim.x + threadIdx.x;  // output-float8 index
    if (tid >= n8) return;

    // tid = plane*32768 + orow*64 + oc8   (512*512/8 per plane, 512/8 per row)
    // input v4f index = plane*16384 + (orow>>1)*64 + oc8
    //                 = ((tid & ~127) >> 1) | (tid & 63)
    unsigned in_idx = ((tid & 0xFFFFFF80u) >> 1) | (tid & 63u);

    v4f a = in4[in_idx];                       // global_load_b128 (RT -> L2)

    v4f o0, o1;
    o0.x = a.x; o0.y = a.x; o0.z = a.y; o0.w = a.y;
    o1.x = a.z; o1.y = a.z; o1.z = a.w; o1.w = a.w;

    unsigned out_idx = tid << 1;               // v4f units
    __builtin_nontemporal_store(o0, out4 + out_idx);      // b128 NT, scale_offset
    __builtin_nontemporal_store(o1, out4 + out_idx + 1);  // b128 NT, ioffset 16
}

extern "C" void kernel_launch(void* const* d_in, const int* in_sizes, int n_in,
                              void* d_out, int out_size, void* d_ws, size_t ws_size,
                              hipStream_t stream) {
    (void)in_sizes; (void)n_in; (void)d_ws; (void)ws_size;
    const v4f* x = (const v4f*)d_in[0];
    v4f* y = (v4f*)d_out;

    // out_size = 8*64*512*512 = 134,217,728 (divisible by 8).
    unsigned n8 = (unsigned)(out_size >> 3);
    unsigned threads = 256;
    unsigned blocks  = (n8 + threads - 1) / threads;  // 65536

    upsample2x_nn_f32_kernel<<<blocks, threads, 0, stream>>>(x, y, n8);
}